// TwinTTL_77077483094255
// MI455X (gfx1250) — compile-verified
//
#include <hip/hip_runtime.h>
#include <math.h>

#define BATCH 32768
#define LRATE 0.01f
#define AEPS  1e-8f
#define LNEPS 1e-5f

typedef __attribute__((ext_vector_type(2))) float v2f;
typedef __attribute__((ext_vector_type(8))) float v8f;

__device__ __forceinline__ v2f mkv2(float x, float y) { v2f v; v[0] = x; v[1] = y; return v; }

__device__ __forceinline__ v8f wmma_f32(v2f a, v2f b, v8f c) {
  // D(16x16,f32) = A(16x4,f32) * B(4x16,f32) + C ; exact fp32 path on CDNA5
  return __builtin_amdgcn_wmma_f32_16x16x4_f32(false, a, false, b, (short)0, c, false, false);
}

// ---- workspace layout (float offsets) ----
constexpr size_t F_CAT  = 0;                                 // B x 16 (K-padded cat)
constexpr size_t F_S    = F_CAT + (size_t)BATCH * 16;        // 3B x 64 (q|k|v rows)
constexpr size_t F_W1   = F_S + (size_t)3 * BATCH * 64;      // 16 x 64  fc1^T padded
constexpr size_t F_HB   = F_W1 + 16 * 64;                    // 64       fc1_b + pe0
constexpr size_t F_QT   = F_HB + 64;                         // 64 x 64  q_w^T
constexpr size_t F_KT   = F_QT + 64 * 64;                    // 64 x 64  k_w^T
constexpr size_t F_VT   = F_KT + 64 * 64;                    // 64 x 64  v_w^T
constexpr size_t F_LWT  = F_VT + 64 * 64;                    // 64 x 64  lw^T
constexpr size_t F_GW   = F_LWT + 64 * 64;                   // 64 x 64  grad W accum
constexpr size_t F_GB   = F_GW + 64 * 64;                    // 64       grad b accum
constexpr size_t F_LW1T = F_GB + 64;                         // 64 x 64  lw1^T
constexpr size_t F_LB1  = F_LW1T + 64 * 64;                  // 64       lb1

// ---------------------------------------------------------------------------
// Kernel 1: streaming reductions -> cat[B][16]. One wave per batch row.
// Bandwidth-bound stage (~315MB), weights staged in LDS.
// ---------------------------------------------------------------------------
__global__ void __launch_bounds__(256) k_cat(
    const float* __restrict__ in1, const float* __restrict__ in2,
    const float* __restrict__ in3, const float* __restrict__ in4,
    const float* __restrict__ wl, const float* __restrict__ bl,
    const float* __restrict__ wm, const float* __restrict__ bm,
    const float* __restrict__ wsw, const float* __restrict__ bs,
    float* __restrict__ cat)
{
  __shared__ float sw[4500 + 2250 + 450];
  for (int i = threadIdx.x; i < 7200; i += 256) {
    float v;
    if (i < 4500)      v = wl[i];
    else if (i < 6750) v = wm[i - 4500];
    else               v = wsw[i - 6750];
    sw[i] = v;
  }
  __syncthreads();

  const int warp = threadIdx.x >> 5, lane = threadIdx.x & 31;
  const int b = blockIdx.x * 8 + warp;

  float aL[3] = {0.f, 0.f, 0.f}, aM[3] = {0.f, 0.f, 0.f}, aS[3] = {0.f, 0.f, 0.f};

  const float* r1 = in1 + (size_t)b * 1500;
  for (int k = lane; k < 500; k += 32) {
    __builtin_prefetch(r1 + (k + 128) * 3, 0, 0);
    float x0 = r1[k * 3 + 0], x1 = r1[k * 3 + 1], x2 = r1[k * 3 + 2];
#pragma unroll
    for (int o = 0; o < 3; ++o)
      aL[o] += x0 * sw[(o * 3 + 0) * 500 + k] + x1 * sw[(o * 3 + 1) * 500 + k] +
               x2 * sw[(o * 3 + 2) * 500 + k];
  }
  const float* r2 = in2 + (size_t)b * 750;
  for (int k = lane; k < 250; k += 32) {
    float x0 = r2[k * 3 + 0], x1 = r2[k * 3 + 1], x2 = r2[k * 3 + 2];
#pragma unroll
    for (int o = 0; o < 3; ++o)
      aM[o] += x0 * sw[4500 + (o * 3 + 0) * 250 + k] + x1 * sw[4500 + (o * 3 + 1) * 250 + k] +
               x2 * sw[4500 + (o * 3 + 2) * 250 + k];
  }
  const float* r3 = in3 + (size_t)b * 150;
  for (int k = lane; k < 50; k += 32) {
    float x0 = r3[k * 3 + 0], x1 = r3[k * 3 + 1], x2 = r3[k * 3 + 2];
#pragma unroll
    for (int o = 0; o < 3; ++o)
      aS[o] += x0 * sw[6750 + (o * 3 + 0) * 50 + k] + x1 * sw[6750 + (o * 3 + 1) * 50 + k] +
               x2 * sw[6750 + (o * 3 + 2) * 50 + k];
  }
#pragma unroll
  for (int o = 0; o < 3; ++o)
#pragma unroll
    for (int off = 16; off; off >>= 1) {
      aL[o] += __shfl_xor(aL[o], off, 32);
      aM[o] += __shfl_xor(aM[o], off, 32);
      aS[o] += __shfl_xor(aS[o], off, 32);
    }
  if (lane == 0) {
    float* cr = cat + (size_t)b * 16;
#pragma unroll
    for (int o = 0; o < 3; ++o) {
      cr[o]      = aL[o] + bl[o];
      cr[3 + o]  = aM[o] + bm[o];
      cr[6 + o]  = aS[o] + bs[o];
      cr[9 + o]  = in4[(size_t)b * 3 + o];
      cr[12 + o] = 0.f;
    }
    cr[15] = 0.f;
  }
}

// ---------------------------------------------------------------------------
// Kernel 2: small prep (transposes, padded fc1, bias+pe0, zero accumulators)
// ---------------------------------------------------------------------------
__global__ void k_prep(const float* __restrict__ fc1w, const float* __restrict__ fc1b,
                       const float* __restrict__ qw, const float* __restrict__ kw,
                       const float* __restrict__ vw, const float* __restrict__ lw,
                       float* __restrict__ wsf)
{
  float* W1 = wsf + F_W1; float* hb = wsf + F_HB;
  float* qT = wsf + F_QT; float* kT = wsf + F_KT; float* vT = wsf + F_VT;
  float* lwT = wsf + F_LWT; float* gW = wsf + F_GW; float* gb = wsf + F_GB;
  for (int i = threadIdx.x; i < 1024; i += 256) {
    int k = i >> 6, n = i & 63;
    W1[i] = (k < 12) ? fc1w[n * 12 + k] : 0.f;
  }
  for (int i = threadIdx.x; i < 4096; i += 256) {
    int k = i >> 6, n = i & 63;
    qT[i] = qw[n * 64 + k]; kT[i] = kw[n * 64 + k];
    vT[i] = vw[n * 64 + k]; lwT[i] = lw[n * 64 + k];
    gW[i] = 0.f;
  }
  for (int i = threadIdx.x; i < 64; i += 256) {
    hb[i] = fc1b[i] + ((i & 1) ? 1.0f : 0.0f);  // pe0 = [0,1,0,1,...]
    gb[i] = 0.f;
  }
}

// ---------------------------------------------------------------------------
// Kernel 3: h = cat@W1 + hb; q,k,v = h@{q,k,v}^T + b  -> s rows. WMMA fp32.
// 8 waves/block, one 16-row tile per wave, h staged through padded LDS.
// ---------------------------------------------------------------------------
__global__ void __launch_bounds__(256) k_qkv(
    const float* __restrict__ cat, const float* __restrict__ qb,
    const float* __restrict__ kbv, const float* __restrict__ vb,
    float* __restrict__ s, const float* __restrict__ W1, const float* __restrict__ hb,
    const float* __restrict__ qT, const float* __restrict__ kT, const float* __restrict__ vT)
{
  __shared__ float sh[8][16 * 68];
  const int warp = threadIdx.x >> 5, lane = threadIdx.x & 31;
  const int hi = lane >> 4, nl = lane & 15;
  const int row0 = (blockIdx.x * 8 + warp) * 16;
  float* shw = sh[warp];

  v2f ca[4];
#pragma unroll
  for (int ks = 0; ks < 4; ++ks) {
    int kb = ks * 4 + 2 * hi;
    const float* p = cat + (size_t)(row0 + nl) * 16 + kb;
    ca[ks] = mkv2(p[0], p[1]);
  }
#pragma unroll
  for (int n0 = 0; n0 < 4; ++n0) {
    int n = n0 * 16 + nl;
    v8f c = {};
#pragma unroll
    for (int ks = 0; ks < 4; ++ks) {
      int kb = ks * 4 + 2 * hi;
      c = wmma_f32(ca[ks], mkv2(W1[kb * 64 + n], W1[(kb + 1) * 64 + n]), c);
    }
    float bias = hb[n];
#pragma unroll
    for (int r = 0; r < 8; ++r) shw[(r + 8 * hi) * 68 + n] = c[r] + bias;
  }
  __syncthreads();

  v2f ha[16];
#pragma unroll
  for (int ks = 0; ks < 16; ++ks) {
    int kb = ks * 4 + 2 * hi;
    ha[ks] = mkv2(shw[nl * 68 + kb], shw[nl * 68 + kb + 1]);
  }
  const float* BT[3] = {qT, kT, vT};
  const float* BB[3] = {qb, kbv, vb};
#pragma unroll
  for (int t = 0; t < 3; ++t) {
    float* ob = s + (size_t)t * BATCH * 64;
#pragma unroll
    for (int n0 = 0; n0 < 4; ++n0) {
      int n = n0 * 16 + nl;
      v8f c = {};
#pragma unroll
      for (int ks = 0; ks < 16; ++ks) {
        int kb = ks * 4 + 2 * hi;
        c = wmma_f32(ha[ks], mkv2(BT[t][kb * 64 + n], BT[t][(kb + 1) * 64 + n]), c);
      }
      float bias = BB[t][n];
#pragma unroll
      for (int r = 0; r < 8; ++r)
        ob[(size_t)(row0 + r + 8 * hi) * 64 + n] = c[r] + bias;
    }
  }
}

// ---------------------------------------------------------------------------
// Kernel 4: tv=s@t_k, lv=s@t_v, pred=tv@lw^T+lb, g=c*(pred-lv); gW += g^T@tv,
// gb += sum(g). 4 waves/block, register-resident gW fragments, LDS reduction.
// ---------------------------------------------------------------------------
__global__ void __launch_bounds__(128) k_grad(
    const float* __restrict__ s, const float* __restrict__ tk,
    const float* __restrict__ tvm, const float* __restrict__ lwT,
    const float* __restrict__ lb, float* __restrict__ gW, float* __restrict__ gb)
{
  __shared__ float stv[4][16 * 68];
  __shared__ float sg[4][16 * 68];
  __shared__ float sgW[64 * 64];
  __shared__ float sgb[64];
  for (int i = threadIdx.x; i < 4096; i += 128) sgW[i] = 0.f;
  if (threadIdx.x < 64) sgb[threadIdx.x] = 0.f;
  __syncthreads();

  const int warp = threadIdx.x >> 5, lane = threadIdx.x & 31;
  const int hi = lane >> 4, nl = lane & 15;
  float* tvw = stv[warp];
  float* gw_ = sg[warp];

  const float gscale = 2.0f / (float)(3ull * BATCH * 64);
  const int NT = 3 * BATCH / 16;
  const int wave = blockIdx.x * 4 + warp;
  const int nw = gridDim.x * 4;

  v8f cgw[4][4];
#pragma unroll
  for (int i0 = 0; i0 < 4; ++i0)
#pragma unroll
    for (int j0 = 0; j0 < 4; ++j0) cgw[i0][j0] = (v8f){};

  for (int t = wave; t < NT; t += nw) {
    const int r0 = t * 16;
    v2f sa[16];
#pragma unroll
    for (int ks = 0; ks < 16; ++ks) {
      int kb = ks * 4 + 2 * hi;
      const float* p = s + (size_t)(r0 + nl) * 64 + kb;
      sa[ks] = mkv2(p[0], p[1]);
    }
    // tv and -lv (lv negated into the g staging buffer)
#pragma unroll
    for (int n0 = 0; n0 < 4; ++n0) {
      int n = n0 * 16 + nl;
      v8f ctv = {}, clv = {};
#pragma unroll
      for (int ks = 0; ks < 16; ++ks) {
        int kb = ks * 4 + 2 * hi;
        ctv = wmma_f32(sa[ks], mkv2(tk[kb * 64 + n], tk[(kb + 1) * 64 + n]), ctv);
        clv = wmma_f32(sa[ks], mkv2(tvm[kb * 64 + n], tvm[(kb + 1) * 64 + n]), clv);
      }
#pragma unroll
      for (int r = 0; r < 8; ++r) {
        tvw[(r + 8 * hi) * 68 + n] = ctv[r];
        gw_[(r + 8 * hi) * 68 + n] = -clv[r];
      }
    }
    // pred = tv @ lw^T + lb ; g = gscale*(pred - lv)
#pragma unroll
    for (int n0 = 0; n0 < 4; ++n0) {
      int n = n0 * 16 + nl;
      v8f cp = {};
#pragma unroll
      for (int ks = 0; ks < 16; ++ks) {
        int kb = ks * 4 + 2 * hi;
        v2f a = mkv2(tvw[nl * 68 + kb], tvw[nl * 68 + kb + 1]);
        cp = wmma_f32(a, mkv2(lwT[kb * 64 + n], lwT[(kb + 1) * 64 + n]), cp);
      }
      float lbn = lb[n];
#pragma unroll
      for (int r = 0; r < 8; ++r) {
        int idx = (r + 8 * hi) * 68 + n;
        float g = gscale * (cp[r] + lbn + gw_[idx]);
        gw_[idx] = g;
        atomicAdd(&sgb[n], g);
      }
    }
    // gW += g^T @ tv  (M=i, N=j, K=16 rows)
#pragma unroll
    for (int i0 = 0; i0 < 4; ++i0) {
      int i = i0 * 16 + nl;
#pragma unroll
      for (int j0 = 0; j0 < 4; ++j0) {
        int j = j0 * 16 + nl;
#pragma unroll
        for (int ks = 0; ks < 4; ++ks) {
          int kb = ks * 4 + 2 * hi;
          v2f a = mkv2(gw_[kb * 68 + i], gw_[(kb + 1) * 68 + i]);
          v2f b = mkv2(tvw[kb * 68 + j], tvw[(kb + 1) * 68 + j]);
          cgw[i0][j0] = wmma_f32(a, b, cgw[i0][j0]);
        }
      }
    }
  }
#pragma unroll
  for (int i0 = 0; i0 < 4; ++i0)
#pragma unroll
    for (int j0 = 0; j0 < 4; ++j0)
#pragma unroll
      for (int r = 0; r < 8; ++r)
        atomicAdd(&sgW[(i0 * 16 + r + 8 * hi) * 64 + j0 * 16 + nl], cgw[i0][j0][r]);
  __syncthreads();
  for (int i = threadIdx.x; i < 4096; i += 128) atomicAdd(&gW[i], sgW[i]);
  if (threadIdx.x < 64) atomicAdd(&gb[threadIdx.x], sgb[threadIdx.x]);
}

// ---------------------------------------------------------------------------
// Kernel 5: sign-SGD-like update; store lw1 transposed for the next GEMM.
// ---------------------------------------------------------------------------
__global__ void k_update(const float* __restrict__ lw, const float* __restrict__ lb,
                         const float* __restrict__ gW, const float* __restrict__ gb,
                         float* __restrict__ lw1T, float* __restrict__ lb1)
{
  for (int i = threadIdx.x; i < 4096; i += 256) {
    int n = i >> 6, k = i & 63;
    float g = gW[i];
    lw1T[k * 64 + n] = lw[i] - LRATE * g / (fabsf(g) + AEPS);
  }
  for (int i = threadIdx.x; i < 64; i += 256) {
    float g = gb[i];
    lb1[i] = lb[i] - LRATE * g / (fabsf(g) + AEPS);
  }
}

// ---------------------------------------------------------------------------
// Kernel 6: only v-rows matter. u = sv@t_q; o = u@lw1^T + lb1; LN; SiLU;
// out = a . out_w + out_b. 4 waves/block, one 16-row tile per wave.
// ---------------------------------------------------------------------------
__global__ void __launch_bounds__(128) k_out(
    const float* __restrict__ s, const float* __restrict__ tq,
    const float* __restrict__ lw1T, const float* __restrict__ lb1,
    const float* __restrict__ lng, const float* __restrict__ lnb,
    const float* __restrict__ ow, const float* __restrict__ obp,
    float* __restrict__ out)
{
  __shared__ float su[4][16 * 68];
  __shared__ float so[4][16 * 68];
  const int warp = threadIdx.x >> 5, lane = threadIdx.x & 31;
  const int hi = lane >> 4, nl = lane & 15;
  const int row0 = (blockIdx.x * 4 + warp) * 16;
  float* uw = su[warp];
  float* om = so[warp];
  const float* sv = s + (size_t)2 * BATCH * 64;

  v2f sa[16];
#pragma unroll
  for (int ks = 0; ks < 16; ++ks) {
    int kb = ks * 4 + 2 * hi;
    const float* p = sv + (size_t)(row0 + nl) * 64 + kb;
    sa[ks] = mkv2(p[0], p[1]);
  }
#pragma unroll
  for (int n0 = 0; n0 < 4; ++n0) {
    int n = n0 * 16 + nl;
    v8f c = {};
#pragma unroll
    for (int ks = 0; ks < 16; ++ks) {
      int kb = ks * 4 + 2 * hi;
      c = wmma_f32(sa[ks], mkv2(tq[kb * 64 + n], tq[(kb + 1) * 64 + n]), c);
    }
#pragma unroll
    for (int r = 0; r < 8; ++r) uw[(r + 8 * hi) * 68 + n] = c[r];
  }
#pragma unroll
  for (int n0 = 0; n0 < 4; ++n0) {
    int n = n0 * 16 + nl;
    v8f c = {};
#pragma unroll
    for (int ks = 0; ks < 16; ++ks) {
      int kb = ks * 4 + 2 * hi;
      v2f a = mkv2(uw[nl * 68 + kb], uw[nl * 68 + kb + 1]);
      c = wmma_f32(a, mkv2(lw1T[kb * 64 + n], lw1T[(kb + 1) * 64 + n]), c);
    }
    float bias = lb1[n];
#pragma unroll
    for (int r = 0; r < 8; ++r) om[(r + 8 * hi) * 68 + n] = c[r] + bias;
  }
  __syncthreads();
  if (lane < 16) {
    const float* rowp = om + lane * 68;
    float mu = 0.f;
#pragma unroll
    for (int j = 0; j < 64; ++j) mu += rowp[j];
    mu *= (1.0f / 64.0f);
    float var = 0.f;
#pragma unroll
    for (int j = 0; j < 64; ++j) { float d = rowp[j] - mu; var += d * d; }
    var *= (1.0f / 64.0f);
    float rstd = rsqrtf(var + LNEPS);
    float acc = 0.f;
#pragma unroll
    for (int j = 0; j < 64; ++j) {
      float nrm = (rowp[j] - mu) * rstd * lng[j] + lnb[j];
      float a = nrm / (1.0f + __expf(-nrm));
      acc += a * ow[j];
    }
    out[row0 + lane] = acc + obp[0];
  }
}

// ---------------------------------------------------------------------------
extern "C" void kernel_launch(void* const* d_in, const int* in_sizes, int n_in,
                              void* d_out, int out_size, void* d_ws, size_t ws_size,
                              hipStream_t stream) {
  const float* in1  = (const float*)d_in[0];
  const float* in2  = (const float*)d_in[1];
  const float* in3  = (const float*)d_in[2];
  const float* in4  = (const float*)d_in[3];
  const float* w_l  = (const float*)d_in[4];
  const float* b_l  = (const float*)d_in[5];
  const float* w_m  = (const float*)d_in[6];
  const float* b_m  = (const float*)d_in[7];
  const float* w_s  = (const float*)d_in[8];
  const float* b_s  = (const float*)d_in[9];
  const float* fc1w = (const float*)d_in[10];
  const float* fc1b = (const float*)d_in[11];
  const float* q_w  = (const float*)d_in[12];
  const float* q_b  = (const float*)d_in[13];
  const float* k_w  = (const float*)d_in[14];
  const float* k_b  = (const float*)d_in[15];
  const float* v_w  = (const float*)d_in[16];
  const float* v_b  = (const float*)d_in[17];
  const float* t_k  = (const float*)d_in[18];
  const float* t_v  = (const float*)d_in[19];
  const float* t_q  = (const float*)d_in[20];
  const float* lw   = (const float*)d_in[21];
  const float* lb   = (const float*)d_in[22];
  const float* lng  = (const float*)d_in[23];
  const float* lnb  = (const float*)d_in[24];
  const float* outw = (const float*)d_in[25];
  const float* outb = (const float*)d_in[26];

  float* wsf  = (float*)d_ws;
  float* cat  = wsf + F_CAT;
  float* sbuf = wsf + F_S;
  float* W1   = wsf + F_W1;
  float* hb   = wsf + F_HB;
  float* qT   = wsf + F_QT;
  float* kT   = wsf + F_KT;
  float* vT   = wsf + F_VT;
  float* lwT  = wsf + F_LWT;
  float* gW   = wsf + F_GW;
  float* gb   = wsf + F_GB;
  float* lw1T = wsf + F_LW1T;
  float* lb1  = wsf + F_LB1;

  k_cat<<<BATCH / 8, 256, 0, stream>>>(in1, in2, in3, in4, w_l, b_l, w_m, b_m, w_s, b_s, cat);
  k_prep<<<1, 256, 0, stream>>>(fc1w, fc1b, q_w, k_w, v_w, lw, wsf);
  k_qkv<<<BATCH / 128, 256, 0, stream>>>(cat, q_b, k_b, v_b, sbuf, W1, hb, qT, kT, vT);
  k_grad<<<512, 128, 0, stream>>>(sbuf, t_k, t_v, lwT, lb, gW, gb);
  k_update<<<1, 256, 0, stream>>>(lw, lb, gW, gb, lw1T, lb1);
  k_out<<<BATCH / 64, 128, 0, stream>>>(sbuf, t_q, lw1T, lb1, lng, lnb, outw, outb,
                                        (float*)d_out);
}